// MonDEQLayerFC_4037269258582
// MI455X (gfx1250) — compile-verified
//
#include <hip/hip_runtime.h>
#include <math.h>

#define N_DIM 2048
#define D_DIM 512
#define B_DIM 256
#define ALPHA 0.1f
#define MAX_ITERS 300
#define TOL 1e-4f

// ---- iteration GEMM tiling: 64x64 block tile, 128 threads (4 waves, 2Mx2N) ----
#define BLK_M 64
#define BLK_N 64
#define CHUNK_K 32
#define NCHUNK (N_DIM / CHUNK_K)       // 64
#define SW_LD 34                       // padded LDS stride (even -> b64-aligned, spreads banks)
#define SZ_LD 34
#define GB_X (N_DIM / BLK_M)           // 32
#define GB_Y (B_DIM / BLK_N)           // 4
#define GEMM_BLOCKS (GB_X * GB_Y)      // 128

// ---- buildw tiling: 128x128 block tile, 256 threads (8 waves, 4Mx2N) ----
#define BW_BLK 128
#define BW_CHUNK 32
#define BW_LD 34

typedef __attribute__((ext_vector_type(2))) float v2f;
typedef __attribute__((ext_vector_type(8))) float v8f;

#define WMMA_F32(A, B, C) \
  __builtin_amdgcn_wmma_f32_16x16x4_f32(false, (A), false, (B), (short)0, (C), false, false)

// ---------------- init: z0 = 0, done = 0 ----------------
__global__ __launch_bounds__(256) void init_kernel(float* __restrict__ z0,
                                                   int* __restrict__ done) {
  int idx = blockIdx.x * 256 + threadIdx.x;
  if (idx < N_DIM * B_DIM) z0[idx] = 0.0f;
  if (idx == 0) *done = 0;
}

// ---------------- V' = ALPHA * (U @ x^T + b)  (N x B) ----------------
__global__ __launch_bounds__(256) void vx_kernel(const float* __restrict__ U,
                                                 const float* __restrict__ b,
                                                 const float* __restrict__ x,
                                                 float* __restrict__ Vx) {
  int idx = blockIdx.x * 256 + threadIdx.x;      // idx = i*256 + j
  int i = idx >> 8;
  int j = idx & 255;
  const float4* u4 = (const float4*)(U + (size_t)i * D_DIM);
  const float4* x4 = (const float4*)(x + (size_t)j * D_DIM);
  float s = 0.0f;
#pragma unroll 4
  for (int d = 0; d < D_DIM / 4; ++d) {
    float4 a = u4[d], c = x4[d];
    s += a.x * c.x + a.y * c.y + a.z * c.z + a.w * c.w;
  }
  Vx[idx] = ALPHA * (s + b[i]);
}

// ---------------- W' = ALPHA*((1-m)I - A^T A + S - S^T) + (1-ALPHA) I ----------------
// LDS-tiled A^T A: block computes 128x128 output; per k-chunk stage two
// transposed A panels sAi[i][k], sAj[j][k]; 8 waves (4Mx2N), wave = 32x64.
// Fragment lane maps (fp32 WMMA 16x16x4):
//   A-operand: lane L -> M = L&15 ; VGPR v -> K = v + 2*(L>>4)
//   B-operand: lane L -> N = L&15 ; VGPR v -> K = v + 2*(L>>4)
//   C/D:       lane L, VGPR v -> (M = v + 8*(L>>4), N = L&15)
__global__ __launch_bounds__(256) void buildw_kernel(const float* __restrict__ A,
                                                     const float* __restrict__ S,
                                                     const float* __restrict__ m_raw,
                                                     float* __restrict__ W) {
  const int tid = threadIdx.x;
  const int lane = tid & 31;
  const int wave = tid >> 5;
  const int wm = wave & 3;             // 4 M groups of 32 rows
  const int wn = wave >> 2;            // 2 N groups of 64 cols
  const int lh = lane >> 4;
  const int lc = lane & 15;
  const int i_base = blockIdx.y * BW_BLK;
  const int j_base = blockIdx.x * BW_BLK;

  __shared__ float sAi[BW_BLK * BW_LD];
  __shared__ float sAj[BW_BLK * BW_LD];

  // staging map: 256 threads cover a 32k x 128col panel (transposing)
  const int kr = tid >> 3;             // 0..31 (k row)
  const int cq = (tid & 7) * 4;        // col quad base; + 32*w, w = 0..3

  v8f acc[2][4];
#pragma unroll
  for (int i = 0; i < 2; ++i)
#pragma unroll
    for (int j = 0; j < 4; ++j) acc[i][j] = (v8f){};

  for (int k0 = 0; k0 < N_DIM; k0 += BW_CHUNK) {
    const float* arow = A + (size_t)(k0 + kr) * N_DIM;
#pragma unroll
    for (int w = 0; w < 4; ++w) {
      int cb = cq + 32 * w;
      float4 ai = *(const float4*)(arow + i_base + cb);
      float4 aj = *(const float4*)(arow + j_base + cb);
      sAi[(cb + 0) * BW_LD + kr] = ai.x;
      sAi[(cb + 1) * BW_LD + kr] = ai.y;
      sAi[(cb + 2) * BW_LD + kr] = ai.z;
      sAi[(cb + 3) * BW_LD + kr] = ai.w;
      sAj[(cb + 0) * BW_LD + kr] = aj.x;
      sAj[(cb + 1) * BW_LD + kr] = aj.y;
      sAj[(cb + 2) * BW_LD + kr] = aj.z;
      sAj[(cb + 3) * BW_LD + kr] = aj.w;
    }
    __syncthreads();

#pragma unroll
    for (int kk = 0; kk < BW_CHUNK; kk += 4) {
      const int kl = kk + 2 * lh;
      v2f a0 = *(const v2f*)&sAi[(wm * 32 + lc) * BW_LD + kl];
      v2f a1 = *(const v2f*)&sAi[(wm * 32 + 16 + lc) * BW_LD + kl];
      v2f b0 = *(const v2f*)&sAj[(wn * 64 + 0 + lc) * BW_LD + kl];
      v2f b1 = *(const v2f*)&sAj[(wn * 64 + 16 + lc) * BW_LD + kl];
      v2f b2 = *(const v2f*)&sAj[(wn * 64 + 32 + lc) * BW_LD + kl];
      v2f b3 = *(const v2f*)&sAj[(wn * 64 + 48 + lc) * BW_LD + kl];
      acc[0][0] = WMMA_F32(a0, b0, acc[0][0]);
      acc[0][1] = WMMA_F32(a0, b1, acc[0][1]);
      acc[0][2] = WMMA_F32(a0, b2, acc[0][2]);
      acc[0][3] = WMMA_F32(a0, b3, acc[0][3]);
      acc[1][0] = WMMA_F32(a1, b0, acc[1][0]);
      acc[1][1] = WMMA_F32(a1, b1, acc[1][1]);
      acc[1][2] = WMMA_F32(a1, b2, acc[1][2]);
      acc[1][3] = WMMA_F32(a1, b3, acc[1][3]);
    }
    __syncthreads();
  }

  const float m = log1pf(expf(m_raw[0]));   // softplus
  const float diag = 1.0f - m;
#pragma unroll
  for (int i = 0; i < 2; ++i) {
#pragma unroll
    for (int j = 0; j < 4; ++j) {
#pragma unroll
      for (int v = 0; v < 8; ++v) {
        int row = i_base + wm * 32 + 16 * i + v + 8 * lh;
        int col = j_base + wn * 64 + 16 * j + lc;
        float w = ((row == col) ? diag : 0.0f) - acc[i][j][v]
                  + S[(size_t)row * N_DIM + col] - S[(size_t)col * N_DIM + row];
        // fold FB-splitting step: W' = ALPHA*W + (1-ALPHA)*I
        W[(size_t)row * N_DIM + col] =
            ALPHA * w + ((row == col) ? (1.0f - ALPHA) : 0.0f);
      }
    }
  }
}

// ---------------- one FB-splitting iteration ----------------
// z_out = relu(W' z_in + V'); emits block-partials of ||z_out-z_in||^2, ||z_in||^2.
// 64x64 block tile (128 blocks for occupancy across WGPs), 4 waves (2Mx2N),
// wave tile 32x32 (4 accumulators). Register-prefetch double buffering:
// next k-chunk's global loads overlap the current chunk's WMMA phase.
__global__ __launch_bounds__(128) void iter_kernel(const float* __restrict__ W,
                                                   const float* __restrict__ Vx,
                                                   const float* __restrict__ zin,
                                                   float* __restrict__ zout,
                                                   float* __restrict__ partials,
                                                   const int* __restrict__ done) {
  const int tid = threadIdx.x;
  const int bid = blockIdx.y * GB_X + blockIdx.x;
  const int row_base = blockIdx.x * BLK_M;
  const int col_base = blockIdx.y * BLK_N;

  __shared__ float sW[BLK_M * SW_LD];
  __shared__ float sZ[BLK_N * SZ_LD];
  __shared__ float red[128];

  if (*done) {  // converged: propagate z unchanged (uniform branch)
    for (int e = tid; e < BLK_M * BLK_N; e += 128) {
      int r = row_base + (e >> 6);
      int c = col_base + (e & 63);
      zout[(size_t)r * B_DIM + c] = zin[(size_t)r * B_DIM + c];
    }
    if (tid == 0) { partials[2 * bid] = 0.0f; partials[2 * bid + 1] = 0.0f; }
    return;
  }

  const int lane = tid & 31;
  const int wave = tid >> 5;
  const int wm = wave & 1;             // 2 M groups of 32 rows
  const int wn = wave >> 1;            // 2 N groups of 32 cols
  const int lh = lane >> 4;
  const int lc = lane & 15;

  // staging maps (128 threads)
  const int sw_m = tid >> 1;                 // 0..63 row
  const int sw_k = (tid & 1) * 16;           // 0 or 16; 4 float4 quads each
  const int sz_k = tid >> 2;                 // 0..31 k row
  const int sz_n = (tid & 3) * 4;            // n quad base; + 16*i, i = 0..3

  const float* wbase = W + (size_t)(row_base + sw_m) * N_DIM + sw_k;
  const float* zbase = zin + (size_t)sz_k * B_DIM + col_base;

  v8f acc[2][2];
#pragma unroll
  for (int i = 0; i < 2; ++i)
#pragma unroll
    for (int j = 0; j < 2; ++j) acc[i][j] = (v8f){};

  float4 wreg[4], zreg[4];
  // prologue: fetch chunk 0
#pragma unroll
  for (int q = 0; q < 4; ++q) wreg[q] = *(const float4*)(wbase + 4 * q);
#pragma unroll
  for (int i = 0; i < 4; ++i) zreg[i] = *(const float4*)(zbase + sz_n + 16 * i);
#pragma unroll
  for (int q = 0; q < 4; ++q)
    *(float4*)&sW[sw_m * SW_LD + sw_k + 4 * q] = wreg[q];
#pragma unroll
  for (int i = 0; i < 4; ++i) {
    int nb = sz_n + 16 * i;
    sZ[(nb + 0) * SZ_LD + sz_k] = zreg[i].x;
    sZ[(nb + 1) * SZ_LD + sz_k] = zreg[i].y;
    sZ[(nb + 2) * SZ_LD + sz_k] = zreg[i].z;
    sZ[(nb + 3) * SZ_LD + sz_k] = zreg[i].w;
  }
  __syncthreads();

  for (int c = 0; c < NCHUNK; ++c) {
    // prefetch chunk c+1 into registers (overlaps WMMA phase below)
    if (c + 1 < NCHUNK) {
      const int k1 = (c + 1) * CHUNK_K;
#pragma unroll
      for (int q = 0; q < 4; ++q) wreg[q] = *(const float4*)(wbase + k1 + 4 * q);
#pragma unroll
      for (int i = 0; i < 4; ++i)
        zreg[i] = *(const float4*)(zbase + (size_t)k1 * B_DIM + sz_n + 16 * i);
    }

    // compute: 8 k-steps x (2 A-frags + 2 B-frags + 4 WMMA)
#pragma unroll
    for (int kk = 0; kk < CHUNK_K; kk += 4) {
      const int kl = kk + 2 * lh;
      v2f a0 = *(const v2f*)&sW[(wm * 32 + lc) * SW_LD + kl];
      v2f a1 = *(const v2f*)&sW[(wm * 32 + 16 + lc) * SW_LD + kl];
      v2f b0 = *(const v2f*)&sZ[(wn * 32 + lc) * SZ_LD + kl];
      v2f b1 = *(const v2f*)&sZ[(wn * 32 + 16 + lc) * SZ_LD + kl];
      acc[0][0] = WMMA_F32(a0, b0, acc[0][0]);
      acc[0][1] = WMMA_F32(a0, b1, acc[0][1]);
      acc[1][0] = WMMA_F32(a1, b0, acc[1][0]);
      acc[1][1] = WMMA_F32(a1, b1, acc[1][1]);
    }
    __syncthreads();

    if (c + 1 < NCHUNK) {
#pragma unroll
      for (int q = 0; q < 4; ++q)
        *(float4*)&sW[sw_m * SW_LD + sw_k + 4 * q] = wreg[q];
#pragma unroll
      for (int i = 0; i < 4; ++i) {
        int nb = sz_n + 16 * i;
        sZ[(nb + 0) * SZ_LD + sz_k] = zreg[i].x;
        sZ[(nb + 1) * SZ_LD + sz_k] = zreg[i].y;
        sZ[(nb + 2) * SZ_LD + sz_k] = zreg[i].z;
        sZ[(nb + 3) * SZ_LD + sz_k] = zreg[i].w;
      }
      __syncthreads();
    }
  }

  // ---- fused epilogue: relu + convergence partials ----
  float ds = 0.0f, zs = 0.0f;
#pragma unroll
  for (int i = 0; i < 2; ++i) {
#pragma unroll
    for (int j = 0; j < 2; ++j) {
#pragma unroll
      for (int v = 0; v < 8; ++v) {
        int row = row_base + wm * 32 + 16 * i + v + 8 * lh;
        int col = col_base + wn * 32 + 16 * j + lc;
        size_t off = (size_t)row * B_DIM + col;
        float zo = zin[off];
        float zn = fmaxf(acc[i][j][v] + Vx[off], 0.0f);
        zout[off] = zn;
        float d = zn - zo;
        ds += d * d;
        zs += zo * zo;
      }
    }
  }

  // deterministic block tree reduction (no fp atomics)
  red[tid] = ds; __syncthreads();
  for (int s = 64; s > 0; s >>= 1) { if (tid < s) red[tid] += red[tid + s]; __syncthreads(); }
  float dsum = red[0]; __syncthreads();
  red[tid] = zs; __syncthreads();
  for (int s = 64; s > 0; s >>= 1) { if (tid < s) red[tid] += red[tid + s]; __syncthreads(); }
  if (tid == 0) { partials[2 * bid] = dsum; partials[2 * bid + 1] = red[0]; }
}

// ---------------- convergence check ----------------
__global__ void check_kernel(const float* __restrict__ partials,
                             int* __restrict__ done) {
  __shared__ float sd[GEMM_BLOCKS], sz[GEMM_BLOCKS];
  int t = threadIdx.x;
  if (t < GEMM_BLOCKS) { sd[t] = partials[2 * t]; sz[t] = partials[2 * t + 1]; }
  __syncthreads();
  for (int s = GEMM_BLOCKS / 2; s > 0; s >>= 1) {
    if (t < s) { sd[t] += sd[t + s]; sz[t] += sz[t + s]; }
    __syncthreads();
  }
  if (t == 0) {
    float err = sqrtf(sd[0]) / (sqrtf(sz[0]) + 1e-12f);
    if (err < TOL) *done = 1;
  }
}

// ---------------- final copy ----------------
__global__ __launch_bounds__(256) void copy_kernel(const float* __restrict__ src,
                                                   float* __restrict__ dst) {
  int idx = blockIdx.x * 256 + threadIdx.x;
  dst[idx] = src[idx];
}

extern "C" void kernel_launch(void* const* d_in, const int* in_sizes, int n_in,
                              void* d_out, int out_size, void* d_ws, size_t ws_size,
                              hipStream_t stream) {
  const float* A     = (const float*)d_in[0];
  const float* S     = (const float*)d_in[1];
  const float* m_raw = (const float*)d_in[2];
  const float* U     = (const float*)d_in[3];
  const float* b     = (const float*)d_in[4];
  const float* x     = (const float*)d_in[5];

  float* W  = (float*)d_ws;                               // 2048*2048 (pre-folded W')
  float* Vx = W  + (size_t)N_DIM * N_DIM;                 // 2048*256  (pre-folded V')
  float* zA = Vx + (size_t)N_DIM * B_DIM;                 // 2048*256
  float* zB = zA + (size_t)N_DIM * B_DIM;                 // 2048*256
  float* partials = zB + (size_t)N_DIM * B_DIM;           // 2*GEMM_BLOCKS
  int*   done = (int*)(partials + 2 * GEMM_BLOCKS);

  init_kernel<<<(N_DIM * B_DIM) / 256, 256, 0, stream>>>(zA, done);
  buildw_kernel<<<dim3(N_DIM / BW_BLK, N_DIM / BW_BLK), 256, 0, stream>>>(A, S, m_raw, W);
  vx_kernel<<<(N_DIM * B_DIM) / 256, 256, 0, stream>>>(U, b, x, Vx);

  for (int it = 1; it <= MAX_ITERS; ++it) {
    const float* zi = (it & 1) ? zA : zB;
    float*       zo = (it & 1) ? zB : zA;
    iter_kernel<<<dim3(GB_X, GB_Y), 128, 0, stream>>>(W, Vx, zi, zo, partials, done);
    check_kernel<<<1, GEMM_BLOCKS, 0, stream>>>(partials, done);
  }
  // MAX_ITERS is even -> final state lives in zA
  copy_kernel<<<(N_DIM * B_DIM) / 256, 256, 0, stream>>>(zA, (float*)d_out);
}